// MultiBoxLoss_69690139344955
// MI455X (gfx1250) — compile-verified
//
#include <hip/hip_runtime.h>
#include <hip/hip_bf16.h>
#include <float.h>

// Problem constants (SSD300 / COCO multibox loss)
#define BB 64
#define NN 8732
#define CC 81
#define WAVES 8
#define NBLK ((NN + WAVES - 1) / WAVES)   // 1092 blocks along the prior axis

__device__ __forceinline__ float wave_max32(float v) {
#pragma unroll
  for (int o = 16; o > 0; o >>= 1) v = fmaxf(v, __shfl_xor(v, o, 32));
  return v;
}
__device__ __forceinline__ float wave_sum32(float v) {
#pragma unroll
  for (int o = 16; o > 0; o >>= 1) v += __shfl_xor(v, o, 32);
  return v;
}

// Deterministic block-wide sum (fixed reduction order). Result valid on thread 0.
__device__ __forceinline__ float block_sum256(float v, float* red8) {
  const int lane = threadIdx.x & 31, wave = threadIdx.x >> 5;
  v = wave_sum32(v);
  if (lane == 0) red8[wave] = v;
  __syncthreads();
  float r = 0.f;
  if (threadIdx.x == 0) {
#pragma unroll
    for (int w = 0; w < 8; ++w) r += red8[w];
  }
  __syncthreads();  // allow red8 reuse
  return r;
}

// ---------------------------------------------------------------------------
// Kernel 1: per-prior confidence loss (logsumexp - logit[label]) + smooth-L1
// loc loss. One wave32 per prior; lanes cover classes {c, c+32, c+64}.
// Streams 362 MB with NT hints; writes neg_loss (L2-resident) + per-block
// deterministic partials (no FP atomics).
// ---------------------------------------------------------------------------
__global__ __launch_bounds__(256)
void mb_main(const float* __restrict__ pred_conf,
             const float* __restrict__ pred_loc,
             const float* __restrict__ gt_conf,
             const float* __restrict__ gt_loc,
             float* __restrict__ neg_loss,
             float* __restrict__ partials) {
  const int b    = blockIdx.y;
  const int wave = threadIdx.x >> 5;
  const int lane = threadIdx.x & 31;
  const int n    = blockIdx.x * WAVES + wave;

  __shared__ float wacc[WAVES][4];

  float a0 = 0.f, a1 = 0.f, a2 = 0.f, a3 = 0.f;  // posConf, numPos, locSum, numPosLoc
  if (n < NN) {
    const size_t row = (size_t)b * NN + n;
    const float* pc = pred_conf + row * CC;
    const float* gc = gt_conf + row * CC;
    const bool t3 = lane < (CC - 64);  // 17 tail lanes

    // Coalesced, non-temporal: this 362 MB stream must not pollute L2.
    float x0 = __builtin_nontemporal_load(pc + lane);
    float x1 = __builtin_nontemporal_load(pc + lane + 32);
    float g0 = __builtin_nontemporal_load(gc + lane);
    float g1 = __builtin_nontemporal_load(gc + lane + 32);
    float x2 = t3 ? __builtin_nontemporal_load(pc + lane + 64) : -FLT_MAX;
    float g2 = t3 ? __builtin_nontemporal_load(gc + lane + 64) : 0.f;

    float m = wave_max32(fmaxf(fmaxf(x0, x1), x2));
    float e = __expf(x0 - m) + __expf(x1 - m) + (t3 ? __expf(x2 - m) : 0.f);
    float s = wave_sum32(e);
    float d = x0 * g0 + x1 * g1 + (t3 ? x2 * g2 : 0.f);
    float dot = wave_sum32(d);                 // = logit at the one-hot label
    float conf = m + __logf(s) - dot;          // CE against one-hot = lse - x[label]

    if (lane == 0) {
      const bool pos = (g0 == 0.f);            // gt_conf[...,0]==0 -> foreground
      neg_loss[row] = pos ? 0.f : conf;        // candidates for hard-neg mining
      if (pos) { a0 = conf; a1 = 1.f; }
      // smooth-L1 loc loss on the 4 box coords (16 B aligned)
      const float4 pl = *((const float4*)pred_loc + row);
      const float4 gl = *((const float4*)gt_loc + row);
      const float pv[4] = {pl.x, pl.y, pl.z, pl.w};
      const float gv[4] = {gl.x, gl.y, gl.z, gl.w};
      float ls = 0.f, anyf = 0.f;
#pragma unroll
      for (int q = 0; q < 4; ++q) {
        if (gv[q] != 0.f) {
          float dq = fabsf(pv[q] - gv[q]);
          ls += (dq < 1.f) ? 0.5f * dq * dq : (dq - 0.5f);
          anyf = 1.f;
        }
      }
      a2 = ls; a3 = anyf;
    }
  }
  if (lane == 0) { wacc[wave][0] = a0; wacc[wave][1] = a1; wacc[wave][2] = a2; wacc[wave][3] = a3; }
  __syncthreads();
  if (threadIdx.x == 0) {
    float s0 = 0, s1 = 0, s2 = 0, s3 = 0;
#pragma unroll
    for (int w = 0; w < WAVES; ++w) { s0 += wacc[w][0]; s1 += wacc[w][1]; s2 += wacc[w][2]; s3 += wacc[w][3]; }
    float* p = partials + ((size_t)b * NBLK + blockIdx.x) * 4;
    p[0] = s0; p[1] = s1; p[2] = s2; p[3] = s3;
  }
}

// ---------------------------------------------------------------------------
// Kernel 2: per-image hard negative mining. Async-DMA the 8732-float neg_loss
// row into LDS (overlapped with the partial-sum reduction), then exact k-th
// largest via 4-pass radix select on f32 bit patterns (all losses >= 0, so
// uint order == float order), then sum of strictly-greater entries.
// ---------------------------------------------------------------------------
__global__ __launch_bounds__(256)
void mb_select(const float* __restrict__ neg_loss,
               const float* __restrict__ partials,
               float* __restrict__ totals) {
  const int b   = blockIdx.x;
  const int tid = threadIdx.x;

  __shared__ __align__(16) float row[NN];   // 34928 B
  __shared__ unsigned hist[256];
  __shared__ float red8[8];
  __shared__ float sh_stats[4];
  __shared__ unsigned sh_prefix;
  __shared__ int sh_k;

  // --- 1) launch async global->LDS DMA of this image's neg-loss row --------
  {
    const unsigned long long src =
        (unsigned long long)(const void*)(neg_loss + (size_t)b * NN);
    const int chunks = NN / 4;  // 2183 x 16B, NN % 4 == 0
    for (int c = tid; c < chunks; c += blockDim.x) {
      unsigned lds  = (unsigned)(unsigned long long)(&row[c * 4]);  // LDS byte addr
      unsigned voff = (unsigned)(c * 16);                           // byte offset
      asm volatile("global_load_async_to_lds_b128 %0, %1, %2"
                   :: "v"(lds), "v"(voff), "s"(src) : "memory");
    }
  }

  // --- 2) overlap DMA with the per-image partial-sum reduction -------------
  {
    float s0 = 0, s1 = 0, s2 = 0, s3 = 0;
    const float* p = partials + (size_t)b * NBLK * 4;
    for (int j = tid; j < NBLK; j += blockDim.x) {
      s0 += p[j * 4 + 0]; s1 += p[j * 4 + 1]; s2 += p[j * 4 + 2]; s3 += p[j * 4 + 3];
    }
    float r0 = block_sum256(s0, red8);
    float r1 = block_sum256(s1, red8);
    float r2 = block_sum256(s2, red8);
    float r3 = block_sum256(s3, red8);
    if (tid == 0) {
      sh_stats[0] = r0; sh_stats[1] = r1; sh_stats[2] = r2; sh_stats[3] = r3;
      int k = (int)(r1 * 3.0f);              // num_neg = 3 * num_pos, truncated
      if (k < 0) k = 0;
      if (k > NN - 1) k = NN - 1;
      sh_k = k;
    }
  }

  // --- 3) wait for the DMA, make LDS visible to all waves ------------------
  asm volatile("s_wait_asynccnt 0" ::: "memory");
  __syncthreads();

  // --- 4) exact k-th largest (== sorted_desc[k]) via 4x8-bit radix select --
  unsigned prefix = 0;
  for (int shift = 24; shift >= 0; shift -= 8) {
    for (int i = tid; i < 256; i += blockDim.x) hist[i] = 0;
    __syncthreads();
    const unsigned hmask = (shift == 24) ? 0u : (0xFFFFFFFFu << (shift + 8));
    for (int i = tid; i < NN; i += blockDim.x) {
      unsigned v = __float_as_uint(row[i]);
      if ((v & hmask) == prefix) atomicAdd(&hist[(v >> shift) & 255], 1u);
    }
    __syncthreads();
    if (tid == 0) {
      int rem = sh_k, bin = 0;
      for (int t = 255; t >= 0; --t) {
        int c = (int)hist[t];
        if (rem < c) { bin = t; break; }
        rem -= c;
      }
      sh_prefix = prefix | ((unsigned)bin << shift);
      sh_k = rem;
    }
    __syncthreads();
    prefix = sh_prefix;
    __syncthreads();
  }
  const float thresh = __uint_as_float(prefix);  // exact k-th order statistic

  // --- 5) sum of losses strictly above the threshold -----------------------
  float loc = 0.f;
  for (int i = tid; i < NN; i += blockDim.x) {
    float v = row[i];
    if (v > thresh) loc += v;
  }
  float negsum = block_sum256(loc, red8);

  if (tid == 0) {
    const float posconf = sh_stats[0], np = sh_stats[1];
    const float locsum = sh_stats[2], nploc = sh_stats[3];
    totals[b] = posconf / np + negsum / (np * 3.0f) + locsum / nploc;
  }
}

// ---------------------------------------------------------------------------
// Kernel 3: mean over the batch.
// ---------------------------------------------------------------------------
__global__ void mb_final(const float* __restrict__ totals, float* __restrict__ out) {
  if (threadIdx.x == 0 && blockIdx.x == 0) {
    float s = 0.f;
    for (int i = 0; i < BB; ++i) s += totals[i];
    out[0] = s / (float)BB;
  }
}

extern "C" void kernel_launch(void* const* d_in, const int* in_sizes, int n_in,
                              void* d_out, int out_size, void* d_ws, size_t ws_size,
                              hipStream_t stream) {
  const float* pred_conf = (const float*)d_in[0];
  const float* pred_loc  = (const float*)d_in[1];
  const float* gt_conf   = (const float*)d_in[2];
  const float* gt_loc    = (const float*)d_in[3];

  // workspace layout (floats): neg_loss[B*N] | partials[B*NBLK*4] | totals[B]
  float* ws       = (float*)d_ws;
  float* neg_loss = ws;
  float* partials = ws + (size_t)BB * NN;
  float* totals   = partials + (size_t)BB * NBLK * 4;

  dim3 grid1(NBLK, BB);
  mb_main<<<grid1, 256, 0, stream>>>(pred_conf, pred_loc, gt_conf, gt_loc,
                                     neg_loss, partials);
  mb_select<<<BB, 256, 0, stream>>>(neg_loss, partials, totals);
  mb_final<<<1, 64, 0, stream>>>(totals, (float*)d_out);
}